// EdgeDecoderClassify_5214090297366
// MI455X (gfx1250) — compile-verified
//
#include <hip/hip_runtime.h>

typedef __attribute__((ext_vector_type(2))) float v2f;
typedef __attribute__((ext_vector_type(8))) float v8f;

#define A_DIM 13
#define KDIM 200
#define DDIM 256
#define NB 5

// LDS row strides padded for bank-conflict-free 16-lane column reads
#define LL2_STRIDE 204   // 204 % 64 = 12 -> lm*12 mod 64 distinct for lm=0..15
#define LL1_STRIDE 260   // 260 % 64 = 4  -> lm*4  mod 64 distinct for lm=0..15
#define Y_STRIDE   260

__launch_bounds__(256)
__global__ void edge_decoder_fused_kernel(const float* __restrict__ x,
                                          const float* __restrict__ ll1_w,
                                          const float* __restrict__ ll1_b,
                                          const float* __restrict__ ll2_w,
                                          const float* __restrict__ ll2_b,
                                          const float* __restrict__ fl_w,
                                          const float* __restrict__ fl_b,
                                          float* __restrict__ out)
{
    __shared__ float lds_ll2[16 * LL2_STRIDE];   // ll2_w zero-padded to 16 rows
    __shared__ float lds_ll1[16 * LL1_STRIDE];   // ll1_w zero-padded to 16 rows
    __shared__ float lds_Y  [16 * Y_STRIDE];     // Y = ll2w_pad @ x_b  (16 x 256)
    __shared__ float lds_prt[8 * 256];           // stage-2 per-wave partials
    __shared__ float lds_s2w[16];                // sum_k ll2_w[c,k]
    __shared__ float lds_r  [256];               // r[a][c] (16x16)

    const int tid  = threadIdx.x;
    const int wid  = tid >> 5;
    const int lane = tid & 31;
    const int half = lane >> 4;      // which 16-lane half of the wave
    const int lm   = lane & 15;
    const int b    = blockIdx.x;
    const float* __restrict__ xb = x + (size_t)b * (KDIM * DDIM);

    // ---- Stage 0: stage tiny weights into LDS (zero padded rows 13..15) ----
    for (int idx = tid; idx < 16 * KDIM; idx += 256) {
        int c = idx / KDIM, k = idx - c * KDIM;
        lds_ll2[c * LL2_STRIDE + k] = (c < A_DIM) ? ll2_w[c * KDIM + k] : 0.0f;
    }
    for (int idx = tid; idx < 16 * DDIM; idx += 256) {
        int a = idx >> 8, d = idx & 255;
        lds_ll1[a * LL1_STRIDE + d] = (a < A_DIM) ? ll1_w[a * DDIM + d] : 0.0f;
    }
    __syncthreads();

    // s2w[c] = sum_k ll2_w[c,k] (bias folding); rows >= 13 are zero
    if (tid < 16) {
        float s = 0.0f;
        for (int k = 0; k < KDIM; ++k) s += lds_ll2[tid * LL2_STRIDE + k];
        lds_s2w[tid] = s;
    }

    // ---- Stage 1: Y(16x256) = ll2w_pad(16x200) @ x_b(200x256) -------------
    // One M tile; each wave owns 2 of the 16 N-tiles; K = 200 in 50 wmma steps.
    const int nt0 = wid * 2, nt1 = wid * 2 + 1;
    v8f acc0 = {};
    v8f acc1 = {};
    for (int k = 0; k < KDIM / 4; ++k) {
        const int kk = 4 * k + 2 * half;           // this lane-half's K pair
        v2f afrag, bf0, bf1;
        afrag.x = lds_ll2[lm * LL2_STRIDE + kk];
        afrag.y = lds_ll2[lm * LL2_STRIDE + kk + 1];
        bf0.x = xb[(size_t)kk       * DDIM + nt0 * 16 + lm];
        bf0.y = xb[(size_t)(kk + 1) * DDIM + nt0 * 16 + lm];
        bf1.x = xb[(size_t)kk       * DDIM + nt1 * 16 + lm];
        bf1.y = xb[(size_t)(kk + 1) * DDIM + nt1 * 16 + lm];
        acc0 = __builtin_amdgcn_wmma_f32_16x16x4_f32(false, afrag, false, bf0,
                                                     (short)0, acc0, false, false);
        acc1 = __builtin_amdgcn_wmma_f32_16x16x4_f32(false, afrag, false, bf1,
                                                     (short)0, acc1, false, false);
    }
#pragma unroll
    for (int v = 0; v < 8; ++v) {   // C/D layout: row = v + 8*half, col = lm
        lds_Y[(v + 8 * half) * Y_STRIDE + nt0 * 16 + lm] = acc0[v];
        lds_Y[(v + 8 * half) * Y_STRIDE + nt1 * 16 + lm] = acc1[v];
    }
    __syncthreads();

    // ---- Stage 2: rT(16x16) = Y(16x256) @ ll1w_pad^T(256x16) --------------
    // K = 256 split across 8 waves (32 each = 8 wmma steps), reduced via LDS.
    v8f accr = {};
#pragma unroll
    for (int s = 0; s < 8; ++s) {
        const int dd = 4 * (wid * 8 + s) + 2 * half;
        v2f afrag, bfrag;
        afrag.x = lds_Y[lm * Y_STRIDE + dd];
        afrag.y = lds_Y[lm * Y_STRIDE + dd + 1];
        bfrag.x = lds_ll1[lm * LL1_STRIDE + dd];       // B[kk][a=lm] = ll1_w[lm][dd]
        bfrag.y = lds_ll1[lm * LL1_STRIDE + dd + 1];
        accr = __builtin_amdgcn_wmma_f32_16x16x4_f32(false, afrag, false, bfrag,
                                                     (short)0, accr, false, false);
    }
#pragma unroll
    for (int v = 0; v < 8; ++v)
        lds_prt[wid * 256 + (v + 8 * half) * 16 + lm] = accr[v];
    __syncthreads();

    // ---- Reduce partials + fold biases: r[a][c] ---------------------------
    {
        const int c = tid >> 4, a = tid & 15;
        float s = 0.0f;
#pragma unroll
        for (int w = 0; w < 8; ++w) s += lds_prt[w * 256 + c * 16 + a];
        const float b1v = (a < A_DIM) ? ll1_b[a] : 0.0f;
        const float b2v = (c < A_DIM) ? ll2_b[c] : 0.0f;
        lds_r[a * 16 + c] = s + b1v * lds_s2w[c] + b2v;
    }
    __syncthreads();

    // ---- Stage 3: out[b,i,j,n] = fl_w . concat(r[min], r[max]) + fl_b -----
    if (tid < A_DIM * A_DIM) {
        const int i = tid / A_DIM, j = tid - i * A_DIM;
        const int mn = (i < j) ? i : j;
        const int mx = (i < j) ? j : i;
        float* __restrict__ op = out + ((size_t)b * (A_DIM * A_DIM) + tid) * NB;
#pragma unroll
        for (int n = 0; n < NB; ++n) {
            float acc = fl_b[n];
#pragma unroll
            for (int c = 0; c < A_DIM; ++c) {
                acc += fl_w[n * 2 * A_DIM + c]         * lds_r[mn * 16 + c]
                     + fl_w[n * 2 * A_DIM + A_DIM + c] * lds_r[mx * 16 + c];
            }
            op[n] = acc;
        }
    }
}

extern "C" void kernel_launch(void* const* d_in, const int* in_sizes, int n_in,
                              void* d_out, int out_size, void* d_ws, size_t ws_size,
                              hipStream_t stream) {
    const float* x     = (const float*)d_in[0];
    const float* ll1_w = (const float*)d_in[1];
    const float* ll1_b = (const float*)d_in[2];
    const float* ll2_w = (const float*)d_in[3];
    const float* ll2_b = (const float*)d_in[4];
    const float* fl_w  = (const float*)d_in[5];
    const float* fl_b  = (const float*)d_in[6];
    // d_in[7] = max_atoms (compile-time constant A_DIM=13 here)
    float* out = (float*)d_out;

    const int B = 2048;
    edge_decoder_fused_kernel<<<B, 256, 0, stream>>>(x, ll1_w, ll1_b, ll2_w,
                                                     ll2_b, fl_w, fl_b, out);
}